// SparseDualFlow_27212912787565
// MI455X (gfx1250) — compile-verified
//
#include <hip/hip_runtime.h>

typedef float v4f __attribute__((ext_vector_type(4)));
typedef float v2f __attribute__((ext_vector_type(2)));

#define ITERS     20
#define STEPSZ    0.01f
#define MOM       0.9f

#define BLOCK     256
#define F4PT      2                    // float4 chunks per thread per tile
#define TILE4     (BLOCK * F4PT)       // 512 float4 = 8 KB per tile stage
#define MAX_BLOCKS 2048

// ---- CDNA5 async global->LDS copy (GV mode: vdst = LDS byte offset, vaddr = 64-bit VGPR pair) ----
__device__ __forceinline__ void async_load_f4(unsigned lds_off, const v4f* gaddr) {
  asm volatile("global_load_async_to_lds_b128 %0, %1, off"
               :: "v"(lds_off), "v"(gaddr)
               : "memory");
}

// ---- 2-wide recurrence chain: encourages v_pk_fma_f32 selection ----
__device__ __forceinline__ v2f solve2(v2f d) {
  const v2f CA = {MOM * (1.0f - 2.0f * STEPSZ), MOM * (1.0f - 2.0f * STEPSZ)}; // 0.882
  const v2f CF = {2.0f * STEPSZ, 2.0f * STEPSZ};                               // 0.02
  const v2f MS = {-STEPSZ, -STEPSZ};
  const v2f Z  = {0.0f, 0.0f};
  v2f f = Z, a = Z;
  const v2f me = d * MS;               // -0.01*d
  #pragma unroll
  for (int it = 0; it < ITERS; ++it) {
    a = __builtin_elementwise_fma(CA, a, __builtin_elementwise_fma(CF, f, me));
    f = __builtin_elementwise_max(f - a, Z);
  }
  return f;
}

__device__ __forceinline__ v4f solve4(v4f d) {
  v2f lo = {d.x, d.y};
  v2f hi = {d.z, d.w};
  lo = solve2(lo);
  hi = solve2(hi);
  v4f r;
  r.x = lo.x; r.y = lo.y; r.z = hi.x; r.w = hi.y;
  return r;
}

// Main streaming kernel: full tiles only (no bounds checks -> exact ASYNCcnt bookkeeping).
__global__ __launch_bounds__(BLOCK) void sdf_main_kernel(
    const v4f* __restrict__ in, v4f* __restrict__ out, int nTiles) {
  __shared__ v4f smem[2 * TILE4];      // 16 KB double buffer
  const int tid    = (int)threadIdx.x;
  const int stride = (int)gridDim.x;
  const int t0     = (int)blockIdx.x;

  // Prologue: stage first tile into buffer 0.
  if (t0 < nTiles) {
    #pragma unroll
    for (int k = 0; k < F4PT; ++k) {
      const int idx4 = t0 * TILE4 + tid + k * BLOCK;
      async_load_f4((unsigned)(size_t)&smem[tid + k * BLOCK], in + idx4);
    }
  }

  int buf = 0;
  for (int t = t0; t < nTiles; t += stride) {
    const int nt = t + stride;
    if (nt < nTiles) {
      // Stage next tile into the other buffer, then wait until the current
      // tile's F4PT async copies have landed (next tile's stay in flight).
      #pragma unroll
      for (int k = 0; k < F4PT; ++k) {
        const int idx4 = nt * TILE4 + tid + k * BLOCK;
        async_load_f4((unsigned)(size_t)&smem[(buf ^ 1) * TILE4 + tid + k * BLOCK],
                      in + idx4);
      }
      asm volatile("s_wait_asynccnt 2" ::: "memory");
    } else {
      asm volatile("s_wait_asynccnt 0" ::: "memory");
    }

    // L2 prefetch two tiles ahead (global_prefetch_b8).
    const int pf = t + 2 * stride;
    if (pf < nTiles) {
      #pragma unroll
      for (int k = 0; k < F4PT; ++k) {
        const int idx4 = pf * TILE4 + tid + k * BLOCK;
        __builtin_prefetch((const void*)(in + idx4), 0, 1);
      }
    }

    // Consume current tile from LDS, compute 20-step recurrence, NT-store.
    #pragma unroll
    for (int k = 0; k < F4PT; ++k) {
      const int idx4 = t * TILE4 + tid + k * BLOCK;
      v4f d = smem[buf * TILE4 + tid + k * BLOCK];
      v4f r = solve4(d);
      __builtin_nontemporal_store(r, out + idx4);
    }
    buf ^= 1;
  }
}

// Scalar tail kernel for elements not covered by full tiles (unused for n = 2^24).
__global__ __launch_bounds__(BLOCK) void sdf_tail_kernel(
    const float* __restrict__ in, float* __restrict__ out, int start, int n) {
  const int i = start + (int)(blockIdx.x * blockDim.x + threadIdx.x);
  if (i < n) {
    const float CA = MOM * (1.0f - 2.0f * STEPSZ);
    const float CF = 2.0f * STEPSZ;
    float d = in[i];
    float f = 0.0f, a = 0.0f;
    const float me = -STEPSZ * d;
    #pragma unroll
    for (int it = 0; it < ITERS; ++it) {
      a = __builtin_fmaf(CA, a, __builtin_fmaf(CF, f, me));
      f = __builtin_fmaxf(f - a, 0.0f);
    }
    out[i] = f;
  }
}

extern "C" void kernel_launch(void* const* d_in, const int* in_sizes, int n_in,
                              void* d_out, int out_size, void* d_ws, size_t ws_size,
                              hipStream_t stream) {
  (void)n_in; (void)out_size; (void)d_ws; (void)ws_size;
  const float* in = (const float*)d_in[0];
  float* out = (float*)d_out;
  const int n  = in_sizes[0];
  const int n4 = n >> 2;                     // full float4 count
  const int nTiles = n4 / TILE4;             // full tiles

  if (nTiles > 0) {
    const int blocks = nTiles < MAX_BLOCKS ? nTiles : MAX_BLOCKS;
    sdf_main_kernel<<<blocks, BLOCK, 0, stream>>>((const v4f*)in, (v4f*)out, nTiles);
  }
  const int start = nTiles * TILE4 * 4;
  const int rem = n - start;
  if (rem > 0) {
    const int blocks = (rem + BLOCK - 1) / BLOCK;
    sdf_tail_kernel<<<blocks, BLOCK, 0, stream>>>(in, out, start, n);
  }
}